// SKA_48223892800328
// MI455X (gfx1250) — compile-verified
//
#include <hip/hip_runtime.h>

typedef __attribute__((ext_vector_type(4))) unsigned v4u;
typedef __attribute__((ext_vector_type(8))) int      v8i;
typedef __attribute__((ext_vector_type(4))) int      v4i;

constexpr int kB  = 8;
constexpr int kC  = 256;
constexpr int kG  = 8;
constexpr int kGC = kC / kG;   // 32 channels per group
constexpr int kH  = 64;
constexpr int kW  = 64;
constexpr int kK2 = 9;         // 3x3 taps

__global__ __launch_bounds__(256)
void ska_dynconv_kernel(const float* __restrict__ x,
                        const float* __restrict__ w,
                        float* __restrict__ out) {
  // Per-pixel weights for this (b, g, h) row: wS[k][px], k = ky*3+kx
  __shared__ __align__(16) float wS[kK2 * kW];

  const int h   = blockIdx.x;
  const int g   = blockIdx.y;
  const int b   = blockIdx.z;
  const int tid = threadIdx.x;

  // ---- TDM: DMA the 9x64 weight tile w[b,g,:,h,:] into LDS.
  // Wave-uniform guard (scalar branch, no exec masking): only wave 0 issues.
  if (__builtin_amdgcn_readfirstlane(tid) < 32) {
    const unsigned long long gaddr = (unsigned long long)(const void*)
        (w + ((((size_t)b * kG + g) * kK2) * kH + h) * kW);
    const unsigned lds_off = (unsigned)(unsigned long long)
        (__attribute__((address_space(3))) float*)wS;

    v4u g0;
    g0[0] = 1u;                                   // count=1, user descriptor
    g0[1] = lds_off;                              // LDS byte address
    g0[2] = (unsigned)(gaddr & 0xFFFFFFFFull);    // global_addr[31:0]
    g0[3] = (unsigned)((gaddr >> 32) & 0x01FFFFFFull) | (2u << 30); // addr[56:32], type=2

    v8i g1;
    g1[0] = (int)(2u << 16);                      // data_size = 4 bytes
    g1[1] = (int)((unsigned)kW  << 16);           // tensor_dim0 = 64 (bits 63:48)
    g1[2] = (int)((unsigned)kK2 << 16);           // tensor_dim1 = 9  (bits 95:80)
    g1[3] = (int)((unsigned)kW  << 16);           // tile_dim0 = 64   (bits 127:112)
    g1[4] = kK2;                                  // tile_dim1 = 9, tile_dim2 = 0
    g1[5] = kH * kW;                              // tensor_dim0_stride = 4096 elems
    g1[6] = 0;
    g1[7] = 0;

    v4i gz = {0, 0, 0, 0};
#if __clang_major__ >= 23
    v8i gz8 = {0, 0, 0, 0, 0, 0, 0, 0};
    __builtin_amdgcn_tensor_load_to_lds(g0, g1, gz, gz, gz8, 0);
#else
    __builtin_amdgcn_tensor_load_to_lds(g0, g1, gz, gz, 0);
#endif
  }
  __builtin_amdgcn_s_wait_tensorcnt(0);
  __syncthreads();

  const size_t base = (((size_t)b * kC + (size_t)g * kGC) * kH) * kW;
  const float* xg = x + base;      // [c][hh][w], c relative to group
  float*       og = out + base;
  const float4* wS4 = (const float4*)wS;

#pragma unroll
  for (int it = 0; it < 2; ++it) {
    const int idx = tid + it * 256;      // 0..511
    const int c   = idx >> 4;            // 0..31
    const int p   = (idx & 15) << 2;     // 0..60, step 4 (pixel quad)

    // Clamped (always-valid) edge addresses; value zeroed by cndmask below.
    const int il = (p == 0)       ? 0       : p - 1;
    const int ir = (p == kW - 4)  ? kW - 1  : p + 4;

    float4 acc;
    acc.x = acc.y = acc.z = acc.w = 0.f;

#pragma unroll
    for (int ky = 0; ky < 3; ++ky) {
      const int hh = h + ky - 1;           // uniform per block -> scalar branch
      if (hh < 0 || hh >= kH) continue;
      const float* row = xg + ((size_t)c * kH + hh) * kW;

      const float4 xm = *(const float4*)(row + p);   // cols p..p+3
      float xl = row[il];                            // unconditional load
      float xr = row[ir];                            // unconditional load
      xl = (p == 0)      ? 0.f : xl;                 // v_cndmask
      xr = (p == kW - 4) ? 0.f : xr;                 // v_cndmask

      const float4 w0 = wS4[(ky * 3 + 0) * (kW / 4) + (p >> 2)];
      const float4 w1 = wS4[(ky * 3 + 1) * (kW / 4) + (p >> 2)];
      const float4 w2 = wS4[(ky * 3 + 2) * (kW / 4) + (p >> 2)];

      // kx = 0 : x[p+j-1]
      acc.x = fmaf(xl,   w0.x, acc.x);
      acc.y = fmaf(xm.x, w0.y, acc.y);
      acc.z = fmaf(xm.y, w0.z, acc.z);
      acc.w = fmaf(xm.z, w0.w, acc.w);
      // kx = 1 : x[p+j]
      acc.x = fmaf(xm.x, w1.x, acc.x);
      acc.y = fmaf(xm.y, w1.y, acc.y);
      acc.z = fmaf(xm.z, w1.z, acc.z);
      acc.w = fmaf(xm.w, w1.w, acc.w);
      // kx = 2 : x[p+j+1]
      acc.x = fmaf(xm.y, w2.x, acc.x);
      acc.y = fmaf(xm.z, w2.y, acc.y);
      acc.z = fmaf(xm.w, w2.z, acc.z);
      acc.w = fmaf(xr,   w2.w, acc.w);
    }

    *(float4*)(og + ((size_t)c * kH + h) * kW + p) = acc;
  }
}

extern "C" void kernel_launch(void* const* d_in, const int* in_sizes, int n_in,
                              void* d_out, int out_size, void* d_ws, size_t ws_size,
                              hipStream_t stream) {
  const float* x = (const float*)d_in[0];
  const float* w = (const float*)d_in[1];
  float* out = (float*)d_out;

  dim3 grid(kH, kG, kB);   // one block per (h-row, group, batch)
  dim3 block(256);
  hipLaunchKernelGGL(ska_dynconv_kernel, grid, block, 0, stream, x, w, out);
}